// NTM_64579128263297
// MI455X (gfx1250) — compile-verified
//
#include <hip/hip_runtime.h>
#include <math.h>

// Problem constants
constexpr int cB = 64, cD = 256, cH = 512, cN = 16384, cM = 64;
constexpr float cEPS = 1e-4f;

typedef float v2f __attribute__((ext_vector_type(2)));
typedef float v8f __attribute__((ext_vector_type(8)));

#if __has_builtin(__builtin_amdgcn_global_load_async_to_lds_b64) && \
    __has_builtin(__builtin_amdgcn_s_wait_asynccnt)
#define USE_ASYNC_LDS 1
// Builtin parameter type per clang diagnostic: vector of 2 ints (b64 payload)
typedef int v2i_as __attribute__((vector_size(2 * sizeof(int))));
typedef __attribute__((address_space(1))) v2i_as* g_v2i_ptr;
typedef __attribute__((address_space(3))) v2i_as* l_v2i_ptr;
#else
#define USE_ASYNC_LDS 0
#endif

__device__ __forceinline__ float sigm(float x) { return 1.f / (1.f + expf(-x)); }

// ---------------- block reductions (wave32) ----------------
__device__ __forceinline__ float block_sum(float v, float* red) {
  int lane = threadIdx.x & 31, wid = threadIdx.x >> 5;
  #pragma unroll
  for (int o = 16; o > 0; o >>= 1) v += __shfl_down(v, o);
  if (lane == 0) red[wid] = v;
  __syncthreads();
  int nw = blockDim.x >> 5;
  if (wid == 0) {
    v = (lane < nw) ? red[lane] : 0.f;
    #pragma unroll
    for (int o = 16; o > 0; o >>= 1) v += __shfl_down(v, o);
    if (lane == 0) red[0] = v;
  }
  __syncthreads();
  v = red[0];
  __syncthreads();
  return v;
}

__device__ __forceinline__ float block_max(float v, float* red) {
  int lane = threadIdx.x & 31, wid = threadIdx.x >> 5;
  #pragma unroll
  for (int o = 16; o > 0; o >>= 1) v = fmaxf(v, __shfl_down(v, o));
  if (lane == 0) red[wid] = v;
  __syncthreads();
  int nw = blockDim.x >> 5;
  if (wid == 0) {
    v = (lane < nw) ? red[lane] : -INFINITY;
    #pragma unroll
    for (int o = 16; o > 0; o >>= 1) v = fmaxf(v, __shfl_down(v, o));
    if (lane == 0) red[0] = v;
  }
  __syncthreads();
  v = red[0];
  __syncthreads();
  return v;
}

// ---------------- WMMA fp32 GEMM: Out[r,c] = act( X[r,:]·W[c,:] + bias[c] (+Out) ) ----------------
// One wave (32 threads) per 16x16 output tile. Layouts per CDNA5 ISA 7.12.2:
//  A 16x4 f32: lanes 0-15 M=lane {v0:K0, v1:K1}; lanes 16-31 {v0:K2, v1:K3}
//  B 4x16 f32: v0 = rows K0 (lanes0-15) / K2 (lanes16-31); v1 = K1 / K3
//  C/D 16x16 f32: VGPR j -> M=j (lanes0-15), M=j+8 (lanes16-31), N=lane&15
__global__ void ntm_gemm_wmma(const float* __restrict__ X, int ldx,
                              const float* __restrict__ W, int ldw,
                              const float* __restrict__ bias,
                              float* __restrict__ Out, int ldo,
                              int K, int accumulate, int act_tanh) {
  const int r0 = blockIdx.x * 16, c0 = blockIdx.y * 16;
  const int lane = threadIdx.x;          // 0..31, EXEC all ones
  const int half = lane >> 4;            // 0/1 -> K pair select
  const int l16 = lane & 15;
  const float* xrow = X + (size_t)(r0 + l16) * ldx;
  const float* wrow = W + (size_t)(c0 + l16) * ldw;
  v8f acc = {};
  for (int k = 0; k < K; k += 4) {
    int kk = k + 2 * half;
    v2f a, b;
    a[0] = xrow[kk];     a[1] = xrow[kk + 1];
    b[0] = wrow[kk];     b[1] = wrow[kk + 1];
    acc = __builtin_amdgcn_wmma_f32_16x16x4_f32(false, a, false, b,
                                                (short)0, acc, false, false);
  }
  float bv = bias ? bias[c0 + l16] : 0.f;
  #pragma unroll
  for (int j = 0; j < 8; ++j) {
    int row = r0 + j + 8 * half;
    float* o = Out + (size_t)row * ldo + c0 + l16;
    float v = acc[j] + bv;
    if (accumulate) v += *o;
    if (act_tanh) v = tanhf(v);
    *o = v;
  }
}

// ---------------- head params: beta,g,gamma, softmax(s), ||k|| ----------------
__global__ void ntm_head_small(const float* __restrict__ h,
                               const float* __restrict__ Wc, const float* __restrict__ bc,
                               const float* __restrict__ Ws, const float* __restrict__ bs,
                               const float* __restrict__ key,
                               float* __restrict__ params) {
  __shared__ float red[32];
  int b = blockIdx.x, t = threadIdx.x;   // 64 threads
  const float* hr = h + (size_t)b * cH;
  float kv = key[b * cM + t];
  float nk2 = block_sum(kv * kv, red);
  float dots[6];
  for (int o = 0; o < 6; ++o) {
    const float* wrow = (o < 3) ? (Wc + o * cH) : (Ws + (o - 3) * cH);
    float p = 0.f;
    for (int kk = t; kk < cH; kk += 64) p += hr[kk] * wrow[kk];
    dots[o] = block_sum(p, red);
  }
  if (t == 0) {
    float c0 = dots[0] + bc[0], c1 = dots[1] + bc[1], c2 = dots[2] + bc[2];
    float s0 = dots[3] + bs[0], s1 = dots[4] + bs[1], s2 = dots[5] + bs[2];
    float mx = fmaxf(s0, fmaxf(s1, s2));
    float e0 = expf(s0 - mx), e1 = expf(s1 - mx), e2 = expf(s2 - mx);
    float zi = 1.f / (e0 + e1 + e2);
    float* p = params + b * 8;
    p[0] = fmaxf(c0, 0.f) + cEPS;          // beta
    p[1] = sigm(c1);                        // g
    p[2] = fmaxf(c2, 0.f) + 1.f + cEPS;     // gamma
    p[3] = e0 * zi; p[4] = e1 * zi; p[5] = e2 * zi;
    p[6] = sqrtf(nk2) + cEPS;               // ||k|| + eps
  }
}

// ---------------- fused dot + row-norm -> cosine score (streams `memory` once) ----------------
// Tile staged in LDS (stride 66 floats = 264B: 8B-aligned for async b64 writes,
// and (2t+m) mod 64 is unique across a wave's lanes -> conflict-free reads).
__global__ void ntm_score(const float* __restrict__ mem, const float* __restrict__ key,
                          const float* __restrict__ params, float* __restrict__ score) {
  constexpr int ROWS = 128;
  constexpr int STRIDE = 66;
  __shared__ float sm[ROWS * STRIDE];
  __shared__ float sk[cM];
  int b = blockIdx.y, n0 = blockIdx.x * ROWS, t = threadIdx.x; // 128 threads
  if (t < cM) sk[t] = key[b * cM + t];
  const float* gsrc = mem + ((size_t)b * cN + n0) * cM;
#if USE_ASYNC_LDS
  // gfx1250 async data mover: global -> LDS without VGPR round-trip (ASYNCcnt).
  // One b64 moves 2 floats per lane; unit u covers floats [2u, 2u+1] of the tile.
  float* gmut = const_cast<float*>(gsrc);
  for (int u = t; u < ROWS * cM / 2; u += 128) {
    int row = u >> 5;            // 32 units per 64-float row
    int col = (u & 31) * 2;
    __builtin_amdgcn_global_load_async_to_lds_b64(
        (g_v2i_ptr)(gmut + (size_t)u * 2),
        (l_v2i_ptr)&sm[row * STRIDE + col],
        0, 0);
  }
  __builtin_amdgcn_s_wait_asynccnt(0);
  __syncthreads();
#else
  for (int f = t; f < ROWS * cM; f += 128) {
    sm[(f >> 6) * STRIDE + (f & 63)] = gsrc[f];   // fully coalesced
  }
  __syncthreads();
#endif
  const float* row = sm + t * STRIDE;
  float dot = 0.f, ssq = 0.f;
  #pragma unroll
  for (int m = 0; m < cM; ++m) { float v = row[m]; dot += v * sk[m]; ssq += v * v; }
  float beta = params[b * 8 + 0], nk = params[b * 8 + 6];
  float nM = sqrtf(ssq) + cEPS;
  score[(size_t)b * cN + n0 + t] = beta * dot / (nM * nk);
}

// ---------------- softmax -> interpolate -> shift(S=3) -> sharpen -> normalize ----------------
__global__ void ntm_addr(const float* __restrict__ score_g, const float* __restrict__ wprev,
                         const float* __restrict__ params, float* __restrict__ head) {
  __shared__ float red[32];
  const int T = 512;
  int b = blockIdx.x, t = threadIdx.x;
  const float* sc = score_g + (size_t)b * cN;
  const float* wp = wprev + (size_t)b * cN;
  const float* p = params + b * 8;
  float g = p[1], gamma = p[2], s0 = p[3], s1 = p[4], s2 = p[5];
  float lm = -INFINITY;
  for (int i = t; i < cN; i += T) lm = fmaxf(lm, sc[i]);
  float mx = block_max(lm, red);
  float lz = 0.f;
  for (int i = t; i < cN; i += T) lz += expf(sc[i] - mx);
  float invZ = 1.f / block_sum(lz, red);
  auto inter = [&](int i) { return g * expf(sc[i] - mx) * invZ + (1.f - g) * wp[i]; };
  float ls = 0.f;
  for (int i = t; i < cN; i += T) {
    float c = s0 * inter((i + cN - 2) & (cN - 1)) +
              s1 * inter((i + cN - 1) & (cN - 1)) + s2 * inter(i);
    ls += powf(fmaxf(c, 1e-8f), gamma);
  }
  float inv = 1.f / block_sum(ls, red);
  float* hd = head + (size_t)b * cN;
  for (int i = t; i < cN; i += T) {
    float c = s0 * inter((i + cN - 2) & (cN - 1)) +
              s1 * inter((i + cN - 1) & (cN - 1)) + s2 * inter(i);
    hd[i] = powf(fmaxf(c, 1e-8f), gamma) * inv;
  }
}

// ---------------- M_read = head · memory (partials, then reduce; no atomics) ----------------
__global__ void ntm_mread_partial(const float* __restrict__ mem, const float* __restrict__ head,
                                  float* __restrict__ part) {
  int b = blockIdx.y, n0 = blockIdx.x * 256;
  int tx = threadIdx.x, ty = threadIdx.y;  // 64 x 4
  const float* mb = mem + (size_t)b * cN * cM;
  const float* hb = head + (size_t)b * cN;
  float acc = 0.f;
  for (int j = ty; j < 256; j += 4) {
    int n = n0 + j;
    acc += hb[n] * mb[(size_t)n * cM + tx];
  }
  __shared__ float r[4][64];
  r[ty][tx] = acc;
  __syncthreads();
  if (ty == 0)
    part[(size_t)blockIdx.x * (cB * cM) + b * cM + tx] = r[0][tx] + r[1][tx] + r[2][tx] + r[3][tx];
}

__global__ void ntm_mread_reduce(const float* __restrict__ part, float* __restrict__ mread) {
  int t = blockIdx.x * blockDim.x + threadIdx.x;  // 4096
  float s = 0.f;
  for (int c = 0; c < cN / 256; ++c) s += part[(size_t)c * (cB * cM) + t];
  mread[t] = s;
}

// ---------------- x = [inp, M_read] ----------------
__global__ void ntm_build_x(const float* __restrict__ inp, const float* __restrict__ mread,
                            float* __restrict__ x) {
  int b = blockIdx.x, t = threadIdx.x;  // 320
  x[b * (cD + cM) + t] = (t < cD) ? inp[b * cD + t] : mread[b * cM + (t - cD)];
}

// ---------------- LSTM pointwise ----------------
__global__ void ntm_lstm_pw(const float* __restrict__ gates, const float* __restrict__ c_tm1,
                            float* __restrict__ out_h, float* __restrict__ out_c) {
  int idx = blockIdx.x * blockDim.x + threadIdx.x;  // B*H = 32768
  int b = idx >> 9, j = idx & 511;
  const float* gr = gates + (size_t)b * (4 * cH);
  float i_ = gr[j], f_ = gr[cH + j], g_ = gr[2 * cH + j], o_ = gr[3 * cH + j];
  float c = sigm(f_) * c_tm1[idx] + sigm(i_) * tanhf(g_);
  out_c[idx] = c;
  out_h[idx] = sigm(o_) * tanhf(c);
}

// ---------------- M_out = memory*(1 - w*e) + w*a ----------------
__global__ void ntm_memwrite(const float* __restrict__ mem, const float* __restrict__ whead,
                             const float* __restrict__ wa, float* __restrict__ Mout) {
  size_t idx = (size_t)blockIdx.x * 256 + threadIdx.x;
  size_t bn = idx >> 6;
  int m = (int)(idx & 63);
  int b = (int)(bn >> 14);
  float w = whead[bn];
  float e = sigm(wa[b * (2 * cM) + m]);
  float a = wa[b * (2 * cM) + cM + m];
  Mout[idx] = mem[idx] * (1.f - w * e) + w * a;
}

// ---------------- launch ----------------
extern "C" void kernel_launch(void* const* d_in, const int* in_sizes, int n_in,
                              void* d_out, int out_size, void* d_ws, size_t ws_size,
                              hipStream_t stream) {
  const float* inp       = (const float*)d_in[0];
  const float* h_tm1     = (const float*)d_in[1];
  const float* c_tm1     = (const float*)d_in[2];
  const float* read_tm1  = (const float*)d_in[3];
  const float* write_tm1 = (const float*)d_in[4];
  const float* memory    = (const float*)d_in[5];
  const float* W_ih = (const float*)d_in[6];  const float* b_ih = (const float*)d_in[7];
  const float* W_hh = (const float*)d_in[8];  const float* b_hh = (const float*)d_in[9];
  const float* W_rk = (const float*)d_in[10]; const float* b_rk = (const float*)d_in[11];
  const float* W_wk = (const float*)d_in[12]; const float* b_wk = (const float*)d_in[13];
  const float* W_rc = (const float*)d_in[14]; const float* b_rc = (const float*)d_in[15];
  const float* W_wc = (const float*)d_in[16]; const float* b_wc = (const float*)d_in[17];
  const float* W_rs = (const float*)d_in[18]; const float* b_rs = (const float*)d_in[19];
  const float* W_ws = (const float*)d_in[20]; const float* b_ws = (const float*)d_in[21];
  const float* W_w  = (const float*)d_in[22]; const float* b_w  = (const float*)d_in[23];

  float* out = (float*)d_out;
  float* out_h  = out;                        // B*H
  float* out_c  = out + 32768;                // B*H
  float* out_rh = out + 65536;                // B*N
  float* out_wh = out + 65536 + 1048576;      // B*N
  float* out_M  = out + 65536 + 2 * 1048576;  // B*N*M

  float* ws = (float*)d_ws;
  float* key_r = ws;                 // 4096
  float* key_w = ws + 4096;          // 4096
  float* par_r = ws + 8192;          // 512
  float* par_w = ws + 8704;          // 512
  float* mread = ws + 9216;          // 4096
  float* xbuf  = ws + 13312;         // 20480
  float* gates = ws + 33792;         // 131072
  float* wa    = ws + 164864;        // 8192
  float* score = ws + 173056;        // 1048576
  float* part  = ws + 173056 + 1048576; // 262144

  // --- read head ---
  ntm_gemm_wmma<<<dim3(4, 4), 32, 0, stream>>>(h_tm1, cH, W_rk, cH, b_rk, key_r, cM, cH, 0, 1);
  ntm_head_small<<<cB, 64, 0, stream>>>(h_tm1, W_rc, b_rc, W_rs, b_rs, key_r, par_r);
  ntm_score<<<dim3(cN / 128, cB), 128, 0, stream>>>(memory, key_r, par_r, score);
  ntm_addr<<<cB, 512, 0, stream>>>(score, read_tm1, par_r, out_rh);

  // --- M_read ---
  ntm_mread_partial<<<dim3(cN / 256, cB), dim3(64, 4), 0, stream>>>(memory, out_rh, part);
  ntm_mread_reduce<<<16, 256, 0, stream>>>(part, mread);

  // --- LSTM ---
  ntm_build_x<<<cB, cD + cM, 0, stream>>>(inp, mread, xbuf);
  ntm_gemm_wmma<<<dim3(4, 128), 32, 0, stream>>>(xbuf, cD + cM, W_ih, cD + cM, b_ih, gates, 4 * cH, cD + cM, 0, 0);
  ntm_gemm_wmma<<<dim3(4, 128), 32, 0, stream>>>(h_tm1, cH, W_hh, cH, b_hh, gates, 4 * cH, cH, 1, 0);
  ntm_lstm_pw<<<128, 256, 0, stream>>>(gates, c_tm1, out_h, out_c);

  // --- write head (depends on h_t) ---
  ntm_gemm_wmma<<<dim3(4, 4), 32, 0, stream>>>(out_h, cH, W_wk, cH, b_wk, key_w, cM, cH, 0, 1);
  ntm_head_small<<<cB, 64, 0, stream>>>(out_h, W_wc, b_wc, W_ws, b_ws, key_w, par_w);
  ntm_score<<<dim3(cN / 128, cB), 128, 0, stream>>>(memory, key_w, par_w, score);
  ntm_addr<<<cB, 512, 0, stream>>>(score, write_tm1, par_w, out_wh);

  // --- erase/add write ---
  ntm_gemm_wmma<<<dim3(4, 8), 32, 0, stream>>>(out_h, cH, W_w, cH, b_w, wa, 2 * cM, cH, 0, 0);
  ntm_memwrite<<<(cB * cN * cM) / 256, 256, 0, stream>>>(memory, out_wh, wa, out_M);
}